// Attention_72189810311336
// MI455X (gfx1250) — compile-verified
//
#include <hip/hip_runtime.h>
#include <math.h>

typedef __attribute__((ext_vector_type(2))) float v2f;
typedef __attribute__((ext_vector_type(4))) float v4f;
typedef __attribute__((ext_vector_type(8))) float v8f;

constexpr int kB      = 32;
constexpr int kS      = 4096;
constexpr int kH      = 1024;
constexpr int kChunk  = 256;              // rows of S per workgroup
constexpr int kNChunk = kS / kChunk;      // 16 chunks -> 512 workgroups
constexpr int kThreads = 256;             // 8 waves (wave32)
constexpr int kWaves   = kThreads / 32;

// ---------------------------------------------------------------------------
// Kernel 1: per (batch, chunk) -> local softmax partials.
//   hidden[b] staged to LDS via GLOBAL_LOAD_ASYNC_TO_LDS_B128 (ASYNCcnt path).
//   scores via V_WMMA_F32_16X16X4_F32 (GEMV: B-operand = hidden broadcast to
//   all 16 columns, so every D column carries the row score).
//   Then local max/sum + partial context c[h] = sum_s exp(score-m)*enc[s][h].
// ---------------------------------------------------------------------------
__global__ __launch_bounds__(kThreads)
void attn_partial(const float* __restrict__ hidden,
                  const float* __restrict__ enc,
                  float* __restrict__ cpart,   // [B*NCHUNK][H]
                  float* __restrict__ ml)      // [B*NCHUNK][2] = (max, sum)
{
    __shared__ float s_hid[kH];
    __shared__ float s_sc[kChunk];
    __shared__ float s_red[kWaves + 2];

    const int tid  = threadIdx.x;
    const int lane = tid & 31;
    const int wave = tid >> 5;
    const int ksel = lane >> 4;      // 0 -> K=0,1 ; 1 -> K=2,3 (A/B operand halves)
    const int mrow = lane & 15;      // row within 16-row tile (A), col N (C/D)

    const int b     = blockIdx.x / kNChunk;
    const int chunk = blockIdx.x % kNChunk;
    const int s0    = chunk * kChunk;

    // ---- stage hidden[b] into LDS with the async global->LDS engine ----
    // 256 threads x 16B = 4KB; tracked by ASYNCcnt, then block barrier.
    {
        const float* gsrc = hidden + (size_t)b * kH + tid * 4;
        unsigned ldsDst = (unsigned)(size_t)(&s_hid[tid * 4]);
        asm volatile("global_load_async_to_lds_b128 %0, %1, off"
                     :: "v"(ldsDst), "v"(gsrc)
                     : "memory");
        asm volatile("s_wait_asynccnt 0x0" ::: "memory");
    }
    __syncthreads();

    const float* encBase = enc + ((size_t)b * kS + s0) * kH;

    // ---- Phase 1: scores (16 rows per wave per pass, 2 passes) ----
    for (int rg = 0; rg < kChunk / (kWaves * 16); ++rg) {
        const int rowBase = rg * (kWaves * 16) + wave * 16;
        const float* aRow = encBase + (size_t)(rowBase + mrow) * kH + ksel * 2;
        v8f acc = {};
        #pragma unroll 4
        for (int k0 = 0; k0 < kH; k0 += 4) {
            v2f a  = *(const v2f*)(aRow + k0);
            v2f bb = *(const v2f*)(&s_hid[k0 + ksel * 2]);
            // D = A(16x4) * B(4x16, hidden broadcast over N) + C
            acc = __builtin_amdgcn_wmma_f32_16x16x4_f32(
                false, a, false, bb, (short)0, acc, false, false);
        }
        // every N column identical; lanes with N==0 (lane 0 & 16) publish rows
        if (mrow == 0) {
            #pragma unroll
            for (int i = 0; i < 8; ++i)
                s_sc[rowBase + ksel * 8 + i] = acc[i];
        }
    }
    __syncthreads();

    // ---- local softmax over the 256 chunk scores (1 score per thread) ----
    const float myScore = s_sc[tid];
    float m = myScore;
    #pragma unroll
    for (int off = 16; off > 0; off >>= 1)
        m = fmaxf(m, __shfl_xor(m, off));
    if (lane == 0) s_red[wave] = m;
    __syncthreads();
    if (tid == 0) {
        float mm = s_red[0];
        for (int i = 1; i < kWaves; ++i) mm = fmaxf(mm, s_red[i]);
        s_red[kWaves] = mm;
    }
    __syncthreads();
    const float gmax = s_red[kWaves];

    const float p = __expf(myScore - gmax);
    s_sc[tid] = p;
    float l = p;
    #pragma unroll
    for (int off = 16; off > 0; off >>= 1)
        l += __shfl_xor(l, off);
    if (lane == 0) s_red[wave] = l;
    __syncthreads();
    if (tid == 0) {
        float ll = 0.f;
        for (int i = 0; i < kWaves; ++i) ll += s_red[i];
        s_red[kWaves + 1] = ll;
    }
    __syncthreads();
    const float gsum = s_red[kWaves + 1];

    // ---- Phase 2: partial context, thread t owns columns 4t..4t+3 ----
    const int hbase = tid * 4;
    v4f c = {};
    #pragma unroll 4
    for (int s = 0; s < kChunk; ++s) {
        // speculative prefetch 8 rows ahead (global_prefetch_b8); OOB dropped
        __builtin_prefetch(encBase + (size_t)(s + 8) * kH + hbase, 0, 0);
        const float pw = s_sc[s];
        v4f e = *(const v4f*)(encBase + (size_t)s * kH + hbase);
        c += e * pw;
    }

    const int pc = b * kNChunk + chunk;
    *(v4f*)(cpart + (size_t)pc * kH + hbase) = c;
    if (tid == 0) {
        ml[2 * pc]     = gmax;
        ml[2 * pc + 1] = gsum;
    }
}

// ---------------------------------------------------------------------------
// Kernel 2: merge the 16 chunk partials per batch (log-sum-exp rescale).
// ---------------------------------------------------------------------------
__global__ __launch_bounds__(kThreads)
void attn_combine(const float* __restrict__ cpart,
                  const float* __restrict__ ml,
                  float* __restrict__ out)
{
    __shared__ float s_w[kNChunk];
    const int b   = blockIdx.x;
    const int tid = threadIdx.x;

    if (tid == 0) {
        float M = -INFINITY;
        for (int i = 0; i < kNChunk; ++i)
            M = fmaxf(M, ml[2 * (b * kNChunk + i)]);
        float L = 0.f;
        float w[kNChunk];
        for (int i = 0; i < kNChunk; ++i) {
            const float wi = __expf(ml[2 * (b * kNChunk + i)] - M);
            w[i] = wi;
            L += wi * ml[2 * (b * kNChunk + i) + 1];
        }
        const float inv = 1.0f / L;
        for (int i = 0; i < kNChunk; ++i) s_w[i] = w[i] * inv;
    }
    __syncthreads();

    const int hbase = tid * 4;
    v4f c = {};
    #pragma unroll
    for (int i = 0; i < kNChunk; ++i) {
        v4f e = *(const v4f*)(cpart + (size_t)(b * kNChunk + i) * kH + hbase);
        c += e * s_w[i];
    }
    *(v4f*)(out + (size_t)b * kH + hbase) = c;
}

// ---------------------------------------------------------------------------
extern "C" void kernel_launch(void* const* d_in, const int* in_sizes, int n_in,
                              void* d_out, int out_size, void* d_ws, size_t ws_size,
                              hipStream_t stream) {
    const float* hidden = (const float*)d_in[0];          // [B, H]
    const float* enc    = (const float*)d_in[1];          // [B, S, H]
    float*       out    = (float*)d_out;                  // [B, H]

    float* wsf   = (float*)d_ws;
    float* cpart = wsf;                                   // B*NCHUNK*H floats
    float* ml    = wsf + (size_t)kB * kNChunk * kH;       // B*NCHUNK*2 floats

    attn_partial<<<dim3(kB * kNChunk), dim3(kThreads), 0, stream>>>(
        hidden, enc, cpart, ml);
    attn_combine<<<dim3(kB), dim3(kThreads), 0, stream>>>(cpart, ml, out);
}